// AFMM_57166014709855
// MI455X (gfx1250) — compile-verified
//
#include <hip/hip_runtime.h>

// Problem constants
#define B_   8
#define C_   512
#define G_   4
#define D_   128     // channels per group
#define H_   128
#define W_   128
#define HF_  65      // rfft bins along W
#define MTOT_ (H_ * HF_)   // 8320 spatial-frequency points per channel
#define MT_  32            // column tile for the mix GEMM

typedef float v2f __attribute__((ext_vector_type(2)));
typedef float v8f __attribute__((ext_vector_type(8)));

// D(16x16,f32) += A(16x4,f32) x B(4x16,f32)  -- CDNA5 V_WMMA_F32_16X16X4_F32
__device__ __forceinline__ v8f afmm_wmma4(v2f a, v2f b, v8f c) {
  return __builtin_amdgcn_wmma_f32_16x16x4_f32(false, a, false, b, (short)0, c,
                                               false, false);
}

// ---------------------------------------------------------------------------
// 128-point in-LDS radix-2 DIT complex FFT. 128 threads cooperate on one line
// (t = 0..127). sign = -1 forward, +1 inverse. No normalization inside.
// Caller must __syncthreads() after populating re/im.
// ---------------------------------------------------------------------------
__device__ __forceinline__ void afmm_fft128(float* re, float* im, int t, float sign) {
  int rt = (int)(__brev((unsigned)t) >> 25);   // 7-bit reversal
  if (t < rt) {
    float ar = re[t], ai = im[t];
    re[t] = re[rt]; im[t] = im[rt];
    re[rt] = ar;    im[rt] = ai;
  }
  __syncthreads();
  for (int s = 1; s <= 7; ++s) {
    int m = 1 << s, half = m >> 1;
    if (t < 64) {
      int grp = t >> (s - 1);
      int pos = t & (half - 1);
      int i0 = grp * m + pos;
      int i1 = i0 + half;
      float ang = sign * 6.28318530717958647692f * (float)pos / (float)m;
      float sn, cs;
      sincosf(ang, &sn, &cs);
      float ur = re[i0], ui = im[i0];
      float vr = re[i1], vi = im[i1];
      float tr = vr * cs - vi * sn;
      float ti = vr * sn + vi * cs;
      re[i0] = ur + tr; im[i0] = ui + ti;
      re[i1] = ur - tr; im[i1] = ui - ti;
    }
    __syncthreads();
  }
}

// ---------------------------------------------------------------------------
// K1: forward real FFT along W for every (b,c,h) row. Applies the full rfft2
// ortho normalization 1/128 ( = 1/sqrt(H*W) ). 2 rows per 256-thread block.
// ---------------------------------------------------------------------------
__global__ __launch_bounds__(256) void AFMM_fft_w_fwd(
    const float* __restrict__ x, float* __restrict__ sR, float* __restrict__ sI) {
  __shared__ float re[2][W_], im[2][W_];
  int t = threadIdx.x & 127;
  int sel = threadIdx.x >> 7;
  size_t row = (size_t)blockIdx.x * 2 + sel;
  re[sel][t] = x[row * W_ + t];
  im[sel][t] = 0.f;
  __syncthreads();
  afmm_fft128(re[sel], im[sel], t, -1.f);
  if (t < HF_) {
    const float sc = 1.f / 128.f;
    sR[row * HF_ + t] = re[sel][t] * sc;
    sI[row * HF_ + t] = im[sel][t] * sc;
  }
}

// ---------------------------------------------------------------------------
// K2/K4: complex FFT along H (in place) for every (b,c,wf) column.
// sign = -1 forward, +1 inverse. 2 columns per 256-thread block.
// ---------------------------------------------------------------------------
__global__ __launch_bounds__(256) void AFMM_fft_h(
    float* __restrict__ sR, float* __restrict__ sI, float sign) {
  __shared__ float re[2][H_], im[2][H_];
  int t = threadIdx.x & 127;
  int sel = threadIdx.x >> 7;
  int col = blockIdx.x * 2 + sel;
  int bc = col / HF_;
  int wf = col - bc * HF_;
  size_t base = (size_t)bc * MTOT_ + wf;
  re[sel][t] = sR[base + (size_t)t * HF_];
  im[sel][t] = sI[base + (size_t)t * HF_];
  __syncthreads();
  afmm_fft128(re[sel], im[sel], t, sign);
  sR[base + (size_t)t * HF_] = re[sel][t];
  sI[base + (size_t)t * HF_] = im[sel][t];
}

// ---------------------------------------------------------------------------
// K3: grouped complex 2-layer channel MLP + spectrum gating, fully in place.
// Block = one (b, g, 32-column tile). 8 waves x one 16-row WMMA tile = 128 out
// channels. All GEMM math on V_WMMA_F32_16X16X4_F32.
//   A tile (16x4, M=out chan, K=in chan):  lane = M, VGPR0 = K{0,2}, VGPR1 = K{1,3}
//   B tile (4x16, K=in chan, N=column):    mirrored layout from LDS
//   C/D (16x16): VGPR v -> M = v + 8*(lane>=16), N = lane&15
// ---------------------------------------------------------------------------
__global__ __launch_bounds__(256) void AFMM_mix(
    float* __restrict__ sR, float* __restrict__ sI,
    const float* __restrict__ w1, const float* __restrict__ w2,
    const float* __restrict__ b1, const float* __restrict__ b2) {
  __shared__ float xR[D_][MT_], xI[D_][MT_];   // origin spectrum tile
  __shared__ float aR[D_][MT_], aI[D_][MT_];   // layer-1 activations

  const int NMT = MTOT_ / MT_;                 // 260
  int blk = blockIdx.x;
  int mt = blk % NMT;
  int bg = blk / NMT;
  int g = bg % G_;
  int b = bg / G_;
  size_t base = ((size_t)(b * C_ + g * D_)) * MTOT_ + (size_t)mt * MT_;

  int tid = threadIdx.x;
  for (int idx = tid; idx < D_ * MT_; idx += 256) {   // 16 uniform iterations
    int k = idx >> 5;
    int m = idx & 31;
    xR[k][m] = sR[base + (size_t)k * MTOT_ + m];
    xI[k][m] = sI[base + (size_t)k * MTOT_ + m];
  }
  __syncthreads();

  int lane = tid & 31;
  int wave = tid >> 5;
  int hi = lane >> 4;        // half-wave select (K offset for A/B, +8 rows for C/D)
  int ln = lane & 15;
  int obase = wave * 16;     // this wave's 16 output channels

  const float* w1r = w1 + (size_t)(0 * G_ + g) * D_ * D_;
  const float* w1i = w1 + (size_t)(1 * G_ + g) * D_ * D_;
  const float* w2r = w2 + (size_t)(0 * G_ + g) * D_ * D_;
  const float* w2i = w2 + (size_t)(1 * G_ + g) * D_ * D_;
  const float* b1r = b1 + (size_t)(0 * G_ + g) * D_;
  const float* b1i = b1 + (size_t)(1 * G_ + g) * D_;
  const float* b2r = b2 + (size_t)(0 * G_ + g) * D_;
  const float* b2i = b2 + (size_t)(1 * G_ + g) * D_;

  // ---- layer 1: acc = W1 (complex) * X (complex) -------------------------
  v8f acR0 = {}, acI0 = {}, acR1 = {}, acI1 = {};
  for (int k0 = 0; k0 < D_; k0 += 4) {
    int kr = k0 + 2 * hi;
    v2f awr, awi, nwi;
    awr.x = w1r[(size_t)kr * D_ + obase + ln];
    awr.y = w1r[(size_t)(kr + 1) * D_ + obase + ln];
    awi.x = w1i[(size_t)kr * D_ + obase + ln];
    awi.y = w1i[(size_t)(kr + 1) * D_ + obase + ln];
    nwi.x = -awi.x; nwi.y = -awi.y;                   // f32 WMMA: no NEG_A, negate operand
    v2f bR0, bI0, bR1, bI1;
    bR0.x = xR[kr][ln];      bR0.y = xR[kr + 1][ln];
    bI0.x = xI[kr][ln];      bI0.y = xI[kr + 1][ln];
    bR1.x = xR[kr][16 + ln]; bR1.y = xR[kr + 1][16 + ln];
    bI1.x = xI[kr][16 + ln]; bI1.y = xI[kr + 1][16 + ln];
    acR0 = afmm_wmma4(awr, bR0, acR0);
    acR0 = afmm_wmma4(nwi, bI0, acR0);
    acI0 = afmm_wmma4(awr, bI0, acI0);
    acI0 = afmm_wmma4(awi, bR0, acI0);
    acR1 = afmm_wmma4(awr, bR1, acR1);
    acR1 = afmm_wmma4(nwi, bI1, acR1);
    acI1 = afmm_wmma4(awr, bI1, acI1);
    acI1 = afmm_wmma4(awi, bR1, acI1);
  }

  // bias + relu -> LDS for layer 2
#pragma unroll
  for (int v = 0; v < 8; ++v) {
    int o = obase + v + 8 * hi;
    float br = b1r[o], bi = b1i[o];
    aR[o][ln]      = fmaxf(acR0[v] + br, 0.f);
    aI[o][ln]      = fmaxf(acI0[v] + bi, 0.f);
    aR[o][16 + ln] = fmaxf(acR1[v] + br, 0.f);
    aI[o][16 + ln] = fmaxf(acI1[v] + bi, 0.f);
  }
  __syncthreads();

  // ---- layer 2: acc = W2 (complex) * A (complex) -------------------------
  v8f dR0 = {}, dI0 = {}, dR1 = {}, dI1 = {};
  for (int k0 = 0; k0 < D_; k0 += 4) {
    int kr = k0 + 2 * hi;
    v2f awr, awi, nwi;
    awr.x = w2r[(size_t)kr * D_ + obase + ln];
    awr.y = w2r[(size_t)(kr + 1) * D_ + obase + ln];
    awi.x = w2i[(size_t)kr * D_ + obase + ln];
    awi.y = w2i[(size_t)(kr + 1) * D_ + obase + ln];
    nwi.x = -awi.x; nwi.y = -awi.y;
    v2f bR0, bI0, bR1, bI1;
    bR0.x = aR[kr][ln];      bR0.y = aR[kr + 1][ln];
    bI0.x = aI[kr][ln];      bI0.y = aI[kr + 1][ln];
    bR1.x = aR[kr][16 + ln]; bR1.y = aR[kr + 1][16 + ln];
    bI1.x = aI[kr][16 + ln]; bI1.y = aI[kr + 1][16 + ln];
    dR0 = afmm_wmma4(awr, bR0, dR0);
    dR0 = afmm_wmma4(nwi, bI0, dR0);
    dI0 = afmm_wmma4(awr, bI0, dI0);
    dI0 = afmm_wmma4(awi, bR0, dI0);
    dR1 = afmm_wmma4(awr, bR1, dR1);
    dR1 = afmm_wmma4(nwi, bI1, dR1);
    dI1 = afmm_wmma4(awr, bI1, dI1);
    dI1 = afmm_wmma4(awi, bR1, dI1);
  }

  // bias + gate by original spectrum, write back in place
#pragma unroll
  for (int v = 0; v < 8; ++v) {
    int o = obase + v + 8 * hi;
    float br = b2r[o], bi = b2i[o];
    size_t go = base + (size_t)o * MTOT_;
    float pr = dR0[v] + br, pi = dI0[v] + bi;
    float xr = xR[o][ln], xi = xI[o][ln];
    sR[go + ln] = pr * xr - pi * xi;
    sI[go + ln] = pr * xi + pi * xr;
    pr = dR1[v] + br; pi = dI1[v] + bi;
    xr = xR[o][16 + ln]; xi = xI[o][16 + ln];
    sR[go + 16 + ln] = pr * xr - pi * xi;
    sI[go + 16 + ln] = pr * xi + pi * xr;
  }
}

// ---------------------------------------------------------------------------
// K5: Hermitian reconstruction + inverse complex FFT along W, keep real part.
// Applies full irfft2 ortho normalization 1/128.
// ---------------------------------------------------------------------------
__global__ __launch_bounds__(256) void AFMM_ifft_w(
    const float* __restrict__ sR, const float* __restrict__ sI,
    float* __restrict__ out) {
  __shared__ float re[2][W_], im[2][W_];
  int t = threadIdx.x & 127;
  int sel = threadIdx.x >> 7;
  size_t row = (size_t)blockIdx.x * 2 + sel;
  size_t rb = row * HF_;
  if (t < HF_) {
    re[sel][t] = sR[rb + t];
    im[sel][t] = sI[rb + t];
  } else {
    re[sel][t] =  sR[rb + (128 - t)];
    im[sel][t] = -sI[rb + (128 - t)];
  }
  __syncthreads();
  afmm_fft128(re[sel], im[sel], t, 1.f);
  out[row * W_ + t] = re[sel][t] * (1.f / 128.f);
}

// ---------------------------------------------------------------------------
extern "C" void kernel_launch(void* const* d_in, const int* in_sizes, int n_in,
                              void* d_out, int out_size, void* d_ws, size_t ws_size,
                              hipStream_t stream) {
  (void)in_sizes; (void)n_in; (void)out_size; (void)ws_size;
  const float* x  = (const float*)d_in[0];
  const float* w1 = (const float*)d_in[1];
  const float* w2 = (const float*)d_in[2];
  const float* b1 = (const float*)d_in[3];
  const float* b2 = (const float*)d_in[4];
  float* out = (float*)d_out;

  const size_t specN = (size_t)B_ * C_ * H_ * HF_;   // 34,078,720 floats
  float* sR = (float*)d_ws;
  float* sI = sR + specN;                            // ws >= 272.6 MB

  const int rows = B_ * C_ * H_;   // 524288
  const int cols = B_ * C_ * HF_;  // 266240
  const int mixBlocks = B_ * G_ * (MTOT_ / MT_);   // 8320

  AFMM_fft_w_fwd<<<rows / 2, 256, 0, stream>>>(x, sR, sI);
  AFMM_fft_h<<<cols / 2, 256, 0, stream>>>(sR, sI, -1.f);
  AFMM_mix<<<mixBlocks, 256, 0, stream>>>(sR, sI, w1, w2, b1, b2);
  AFMM_fft_h<<<cols / 2, 256, 0, stream>>>(sR, sI, 1.f);
  AFMM_ifft_w<<<rows / 2, 256, 0, stream>>>(sR, sI, out);
}